// SinLSTM_81776177315918
// MI455X (gfx1250) — compile-verified
//
#include <hip/hip_runtime.h>

typedef __attribute__((ext_vector_type(2))) float v2f;
typedef __attribute__((ext_vector_type(4))) float v4f;
typedef __attribute__((ext_vector_type(8))) float v8f;

namespace {
constexpr int HID   = 50;
constexpr int NG    = 200;     // 4*HID gate columns
constexpr int KPAD  = 52;      // 50 hidden + bias row (k=50) + input row (k=51)
constexpr int NT    = 13;      // N tiles of 16 -> 208 padded columns
constexpr int KC    = 13;      // K chunks of 4 -> 52
constexpr int NPAD  = NT * 16; // 208
constexpr int T_IN  = 512;
constexpr int FUT   = 64;
constexpr int T_OUT = T_IN + FUT;

__device__ __forceinline__ float sigmoidf_(float v) {
    return 1.0f / (1.0f + __expf(-v));
}
} // namespace

// One wave (32 lanes) per workgroup; each workgroup owns 16 batch rows and
// runs the entire 576-step recurrence. GEMM per step:
//   gates[16x208] = h_ext[16x52] @ B[52x208]   via 13x13 v_wmma_f32_16x16x4_f32
// where B rows 0..49 = W_hh.T, row 50 = (b_ih+b_hh), row 51 = W_ih, and
// h_ext row has h (k<50), 1.0 (k=50), x_t (k=51).
__global__ __launch_bounds__(32) void lstm_seq_kernel(
    const float* __restrict__ x,      // [B, 512]
    const float* __restrict__ W_ih,   // [200, 1]
    const float* __restrict__ W_hh,   // [200, 50]
    const float* __restrict__ b_ih,   // [200]
    const float* __restrict__ b_hh,   // [200]
    const float* __restrict__ W_out,  // [1, 50]
    const float* __restrict__ b_out,  // [1]
    float* __restrict__ out)          // [B, 576]
{
    // B-operand fragments in WMMA lane layout: [kc][tt][lane] -> float2
    __shared__ float frags[KC * NT * 32 * 2];   // 43264 B
    // gate buffer, column-major [n][r] (n = padded gate column, r = batch row)
    __shared__ float gates[NPAD * 16];          // 13312 B
    // h_ext buffer, row-major [r][k], k in [0,52)
    __shared__ float hbuf[16 * KPAD];           //  3328 B

    const int lane    = threadIdx.x;   // 0..31
    const int r       = lane & 15;     // batch row within tile / column within tile
    const int half    = lane >> 4;     // 0 | 1
    const int rowbase = blockIdx.x * 16;

    // ---------- one-time prestage of B fragments ----------
    // A-layout mirror for 32-bit 4xK operands:
    //   lane half selects K pair {0,1} vs {2,3} within each 4-chunk,
    //   float2 components are the two consecutive K rows; lane%16 = column n.
    for (int fi = 0; fi < KC * NT; ++fi) {
        const int kc = fi / NT;
        const int tt = fi % NT;
        const int n  = tt * 16 + r;
        const int k0 = kc * 4 + 2 * half;
        float v0 = 0.0f, v1 = 0.0f;
        if (n < NG) {
            if (k0 < HID)       v0 = W_hh[n * HID + k0];
            else if (k0 == HID) v0 = b_ih[n] + b_hh[n];
            else                v0 = W_ih[n];
            const int k1 = k0 + 1;
            if (k1 < HID)       v1 = W_hh[n * HID + k1];
            else if (k1 == HID) v1 = b_ih[n] + b_hh[n];
            else                v1 = W_ih[n];
        }
        frags[(fi * 32 + lane) * 2 + 0] = v0;
        frags[(fi * 32 + lane) * 2 + 1] = v1;
    }

    // ---------- per-lane state init ----------
    // Lane owns (row r, k = 2*j + half) for j in [0,25): exactly covers k in [0,50).
    float creg[25];
    float wout[25];
#pragma unroll
    for (int j = 0; j < 25; ++j) {
        const int k = 2 * j + half;
        creg[j] = 0.0f;
        wout[j] = W_out[k];
        hbuf[r * KPAD + k] = 0.0f;      // h0 = 0 (halves write disjoint k)
    }
    hbuf[r * KPAD + HID] = 1.0f;        // constant bias row (dup write, same value)
    const float bo = b_out[0];
    __syncthreads();

    float xnext = 0.0f;

#pragma unroll 1
    for (int t = 0; t < T_OUT; ++t) {
        // input for this step: observed x, then autoregressive output
        float xt;
        if (t < T_IN) xt = x[(rowbase + r) * T_IN + t];
        else          xt = xnext;
        hbuf[r * KPAD + (HID + 1)] = xt;   // input row (both halves: same value)

        // A fragments for this step: lane holds h_ext[r][4kc + 2*half + {0,1}]
        v2f afrag[KC];
#pragma unroll
        for (int kc = 0; kc < KC; ++kc)
            afrag[kc] = *(const v2f*)&hbuf[r * KPAD + kc * 4 + 2 * half];

        // gates = h_ext @ B  (16x52 @ 52x208), 169 fp32 WMMAs
#pragma unroll
        for (int tt = 0; tt < NT; ++tt) {
            v8f acc = {0.f, 0.f, 0.f, 0.f, 0.f, 0.f, 0.f, 0.f};
#pragma unroll
            for (int kc = 0; kc < KC; ++kc) {
                const v2f bfrag =
                    *(const v2f*)&frags[((kc * NT + tt) * 32 + lane) * 2];
                acc = __builtin_amdgcn_wmma_f32_16x16x4_f32(
                    /*neg_a=*/false, afrag[kc],
                    /*neg_b=*/false, bfrag,
                    /*c_mod=*/(short)0, acc,
                    /*reuse_a=*/false, /*reuse_b=*/false);
            }
            // D layout: VGPR v holds row (v + 8*half), column n = tt*16 + r.
            const int n = tt * 16 + r;
            float* gp = &gates[n * 16 + half * 8];
            v4f lo = {acc[0], acc[1], acc[2], acc[3]};
            v4f hi = {acc[4], acc[5], acc[6], acc[7]};
            *(v4f*)(gp)     = lo;
            *(v4f*)(gp + 4) = hi;
        }

        // elementwise LSTM cell + output projection (lane: row r, 25 k values)
        float osum = 0.0f;
#pragma unroll
        for (int j = 0; j < 25; ++j) {
            const int k = 2 * j + half;
            const float gi = gates[(k          ) * 16 + r];
            const float gf = gates[(k +     HID) * 16 + r];
            const float gg = gates[(k + 2 * HID) * 16 + r];
            const float go = gates[(k + 3 * HID) * 16 + r];
            const float ig = sigmoidf_(gi);
            const float fg = sigmoidf_(gf);
            const float cg = tanhf(gg);
            const float og = sigmoidf_(go);
            const float cv = fg * creg[j] + ig * cg;
            creg[j] = cv;
            const float hv = og * tanhf(cv);
            osum += hv * wout[j];
            hbuf[r * KPAD + k] = hv;       // h for next step's A operand
        }

        // reduce the two half-wave partial sums (k-even + k-odd) for row r;
        // after the xor-shuffle both halves hold the full projection.
        osum += __shfl_xor(osum, 16, 32);
        const float ov = osum + bo;
        if (half == 0)
            out[(rowbase + r) * T_OUT + t] = ov;
        xnext = ov;   // autoregressive input for the future phase
    }
}

extern "C" void kernel_launch(void* const* d_in, const int* in_sizes, int n_in,
                              void* d_out, int out_size, void* d_ws, size_t ws_size,
                              hipStream_t stream) {
    const float* x     = (const float*)d_in[0];
    const float* W_ih  = (const float*)d_in[1];
    const float* W_hh  = (const float*)d_in[2];
    const float* b_ih  = (const float*)d_in[3];
    const float* b_hh  = (const float*)d_in[4];
    const float* W_out = (const float*)d_in[5];
    const float* b_out = (const float*)d_in[6];
    float* out = (float*)d_out;

    const int B = in_sizes[0] / T_IN;   // 2048
    dim3 grid(B / 16);                  // 128 single-wave workgroups
    dim3 block(32);
    hipLaunchKernelGGL(lstm_seq_kernel, grid, block, 0, stream,
                       x, W_ih, W_hh, b_ih, b_hh, W_out, b_out, out);
}